// S_Att_Pair_Bias_41583873360516
// MI455X (gfx1250) — compile-verified
//
#include <hip/hip_runtime.h>
#include <math.h>

#define L_DIM 1024
#define SEQ_DIM 384
#define PAIR_DIM 128
#define HEADS 12
#define HIDDEN 32
#define EPS 1e-5f

typedef float v2f __attribute__((ext_vector_type(2)));
typedef float v8f __attribute__((ext_vector_type(8)));
typedef int   v4i __attribute__((ext_vector_type(4)));
typedef __attribute__((address_space(1))) v4i* as1_v4i;
typedef __attribute__((address_space(3))) v4i* as3_v4i;

#if __has_builtin(__builtin_amdgcn_global_load_async_to_lds_b128)
#define ASYNC_LDS 1
#endif

// D(16x16,f32) += A(16x4,f32) * B(4x16,f32) on one wave32.
// A: lane l<16 holds row l, K={0,1}; lane l+16 holds row l, K={2,3}.
// B: lane n<16 holds col n, K={0,1}; lane n+16 holds col n, K={2,3}.
// D: vgpr r, lanes 0-15 -> row r, lanes 16-31 -> row r+8; lane&15 = col.
static __device__ __forceinline__ v8f wmma4(float a0, float a1, float b0, float b1, v8f c) {
  v2f a = {a0, a1};
  v2f b = {b0, b1};
  return __builtin_amdgcn_wmma_f32_16x16x4_f32(false, a, false, b, (short)0, c, false, false);
}

// ---------------------------------------------------------------------------
// 1. LayerNorm of seq_embed [1024, 384]; one wave per row.
// ---------------------------------------------------------------------------
__global__ void ln_seq_kernel(const float* __restrict__ x, const float* __restrict__ gamma,
                              const float* __restrict__ beta, float* __restrict__ out) {
  int row = blockIdx.x;
  int l = threadIdx.x;
  const float* xr = x + (size_t)row * SEQ_DIM;
  float v[12];
  float s = 0.f, sq = 0.f;
#pragma unroll
  for (int t = 0; t < 12; ++t) {
    float f = xr[l + 32 * t];
    v[t] = f; s += f; sq += f * f;
  }
#pragma unroll
  for (int m = 16; m >= 1; m >>= 1) {
    s  += __shfl_xor(s,  m, 32);
    sq += __shfl_xor(sq, m, 32);
  }
  float mu  = s * (1.f / SEQ_DIM);
  float var = sq * (1.f / SEQ_DIM) - mu * mu;
  float rs  = rsqrtf(var + EPS);
  float* orow = out + (size_t)row * SEQ_DIM;
#pragma unroll
  for (int t = 0; t < 12; ++t) {
    int idx = l + 32 * t;
    orow[idx] = (v[t] - mu) * rs * gamma[idx] + beta[idx];
  }
}

// ---------------------------------------------------------------------------
// Generic 16x16 fp32 WMMA tile: C = A[16,K] * W[16,K]^T (W rows are B cols).
// ---------------------------------------------------------------------------
static __device__ __forceinline__ v8f gemm_tile_f32(const float* __restrict__ A, int lda,
                                                    const float* __restrict__ W, int ldw, int K) {
  int l = threadIdx.x & 31;
  int low = l & 15;
  int koff = (l >> 4) * 2;
  const float* ap = A + (size_t)low * lda + koff;
  const float* wp = W + (size_t)low * ldw + koff;
  v8f acc = {};
#pragma unroll 8
  for (int kk = 0; kk < K; kk += 4) {
    float2 a = *(const float2*)(ap + kk);
    float2 b = *(const float2*)(wp + kk);
    acc = wmma4(a.x, a.y, b.x, b.y, acc);
  }
  return acc;
}

// ---------------------------------------------------------------------------
// 2. Q/K/V/Gate projections: [1024,384] @ W^T, stored per-head [h][i][32].
//    grid = (64 Mtiles, 24 Ntiles, 4 matrices), block = 1 wave.
// ---------------------------------------------------------------------------
__global__ void proj_qkvg_kernel(const float* __restrict__ s_norm,
                                 const float* __restrict__ Wq, const float* __restrict__ Wk,
                                 const float* __restrict__ Wv, const float* __restrict__ Wg,
                                 const float* __restrict__ bg,
                                 float* __restrict__ q, float* __restrict__ k,
                                 float* __restrict__ v, float* __restrict__ g) {
  int i0 = blockIdx.x * 16, n0 = blockIdx.y * 16, which = blockIdx.z;
  const float* W = (which == 0) ? Wq : (which == 1) ? Wk : (which == 2) ? Wv : Wg;
  float* dst     = (which == 0) ? q  : (which == 1) ? k  : (which == 2) ? v  : g;
  v8f acc = gemm_tile_f32(s_norm + (size_t)i0 * SEQ_DIM, SEQ_DIM,
                          W + (size_t)n0 * SEQ_DIM, SEQ_DIM, SEQ_DIM);
  int l = threadIdx.x & 31, low = l & 15, hi = l >> 4;
  int c = n0 + low, h = c >> 5, d = c & 31;
#pragma unroll
  for (int r = 0; r < 8; ++r) {
    int i = i0 + r + 8 * hi;
    float val = acc[r];
    if (which == 0)      val *= 0.17677669529663688f;                        // 1/sqrt(32)
    else if (which == 3) val = __builtin_amdgcn_rcpf(1.f + __expf(-(val + bg[c]))); // gate
    dst[((size_t)h * L_DIM + i) * HIDDEN + d] = val;
  }
}

// ---------------------------------------------------------------------------
// 3. Fused pair LayerNorm + bias projection (the 512 MB kernel).
//    bias_h(i,j) = rsig*(z.(gamma*Wb_h) - mu*S_h) + C_h, done as a raw GEMM:
//    A = z rows (16 i's, fixed j per wave), B = gamma*Wb padded to 16 cols
//    (col 12 = ones -> row sums for mu). Sumsq folded into the A load.
//    The block's 16x1024-float z region is staged Global->LDS with
//    GLOBAL_LOAD_ASYNC_TO_LDS_B128 (fully coalesced 512B per instruction),
//    and the WMMA loop reads A from LDS bank-conflict-free (pitch 1028).
//    Output written transposed per head: bias[h][j][i] (coalesced via LDS).
//    grid = (128 j-blocks of 8, 64 i-tiles), block = 256 (8 waves).
// ---------------------------------------------------------------------------
#define ZPITCH 1028   // 1024 + 4 pad: lanes hit banks 4*low(+2hi), disjoint
__global__ void pair_bias_kernel(const float* __restrict__ z,
                                 const float* __restrict__ zg, const float* __restrict__ zb,
                                 const float* __restrict__ Wb, float* __restrict__ bias) {
  __shared__ float  zsh[16 * ZPITCH]; // 64.25 KB staged A tile
  __shared__ float2 Bsh[1024];        // [step 0..31][lane 0..31]
  __shared__ float  Ssh[16], Csh[16];
  __shared__ float  stage[12 * 137];  // [h]*137 + [j_l]*17 + [i_l]
  int tid = threadIdx.x;
  int i0 = blockIdx.y * 16;

  // Issue async copy of z[i0..i0+15][j0..j0+7][:] into LDS (16 rows x 4KB).
  {
    const float* gbase = z + ((size_t)i0 * L_DIM + (size_t)blockIdx.x * 8) * PAIR_DIM;
#pragma unroll
    for (int r = 0; r < 16; ++r) {
      const float* gsrc = gbase + (size_t)r * L_DIM * PAIR_DIM + tid * 4;
      float* ldst = &zsh[r * ZPITCH + tid * 4];
#ifdef ASYNC_LDS
      __builtin_amdgcn_global_load_async_to_lds_b128(
          (as1_v4i)(void*)gsrc, (as3_v4i)(void*)ldst, 0, 0);
#else
      *(float4*)ldst = *(const float4*)gsrc;
#endif
    }
  }

  // Meanwhile build B tile (gamma-fused Wb, ones column), plus S_h / C_h.
  for (int idx = tid; idx < 1024; idx += 256) {
    int step = idx >> 5, ln = idx & 31;
    int n = ln & 15, p = step * 4 + 2 * (ln >> 4);
    float fx, fy;
    if (n < 12)       { fx = zg[p] * Wb[n * PAIR_DIM + p]; fy = zg[p + 1] * Wb[n * PAIR_DIM + p + 1]; }
    else if (n == 12) { fx = 1.f; fy = 1.f; }
    else              { fx = 0.f; fy = 0.f; }
    Bsh[idx] = make_float2(fx, fy);
  }
  if (tid < 16) {
    float s = 0.f, c = 0.f;
    if (tid < 12) {
      const float* wr = Wb + tid * PAIR_DIM;
      for (int p = 0; p < PAIR_DIM; ++p) { s += zg[p] * wr[p]; c += zb[p] * wr[p]; }
    }
    Ssh[tid] = s; Csh[tid] = c;
  }

#ifdef ASYNC_LDS
#if __has_builtin(__builtin_amdgcn_s_wait_asynccnt)
  __builtin_amdgcn_s_wait_asynccnt(0);
#else
  asm volatile("s_wait_asynccnt 0x0" ::: "memory");
#endif
#endif
  __syncthreads();

  int w = tid >> 5, l = tid & 31, low = l & 15, hi = l >> 4;
  const float* arow = &zsh[low * ZPITCH + w * PAIR_DIM + 2 * hi];

  float2 barr[32];
#pragma unroll
  for (int t = 0; t < 32; ++t) barr[t] = Bsh[t * 32 + l];

  v8f acc = {};
  float sq = 0.f;
#pragma unroll
  for (int t = 0; t < 32; ++t) {
    float2 a = *(const float2*)(arow + 4 * t);
    sq = fmaf(a.x, a.x, sq);
    sq = fmaf(a.y, a.y, sq);
    acc = wmma4(a.x, a.y, barr[t].x, barr[t].y, acc);
  }
  sq += __shfl_xor(sq, 16, 32);   // lanes l and l+16 cover disjoint halves of a row

  float Sh = Ssh[low], Ch = Csh[low];
#pragma unroll
  for (int r = 0; r < 8; ++r) {
    float rowsum = __shfl(acc[r], 12 + (l & 16), 32);          // ones-column -> sum_p z_p
    float sqr    = __shfl(sq, r + ((l & 16) >> 1), 32);        // row (r+8*hi)'s sumsq
    float mu   = rowsum * (1.f / PAIR_DIM);
    float var  = sqr * (1.f / PAIR_DIM) - mu * mu;
    float rsig = rsqrtf(var + EPS);
    float val  = rsig * (acc[r] - mu * Sh) + Ch;
    if (low < 12) stage[low * 137 + w * 17 + (r + 8 * hi)] = val;
  }
  __syncthreads();

  size_t j0 = (size_t)blockIdx.x * 8;
  for (int idx = tid; idx < 12 * 8 * 16; idx += 256) {
    int h = idx >> 7, rem = idx & 127, jl = rem >> 4, il = rem & 15;
    bias[(size_t)h * L_DIM * L_DIM + (j0 + jl) * L_DIM + i0 + il] =
        stage[h * 137 + jl * 17 + il];
  }
}

// ---------------------------------------------------------------------------
// 4. Flash attention with pair bias and gating.
//    Per block: one (head, 16-query tile); 4 waves split the j range.
//    Computes T = S^T = K*Q^T + bias^T so online softmax is per-lane,
//    and O^T = V^T * P so rescaling is per-lane. grid=(64,12), block=128.
// ---------------------------------------------------------------------------
__global__ void attention_kernel(const float* __restrict__ q, const float* __restrict__ k,
                                 const float* __restrict__ v, const float* __restrict__ g,
                                 const float* __restrict__ bias, float* __restrict__ op) {
  __shared__ float m_s[4][16], s_s[4][16], acc_s[4][32][16];
  int h = blockIdx.y, i0 = blockIdx.x * 16;
  int tid = threadIdx.x, w = tid >> 5, l = tid & 31, low = l & 15, hi = l >> 4;
  const float* qh = q + (size_t)h * L_DIM * HIDDEN;
  const float* kh = k + (size_t)h * L_DIM * HIDDEN;
  const float* vh = v + (size_t)h * L_DIM * HIDDEN;
  const float* bh = bias + (size_t)h * L_DIM * L_DIM;

  float2 qB[8];   // B operand (Q) is tile-invariant: preload 8 K-steps
#pragma unroll
  for (int t = 0; t < 8; ++t)
    qB[t] = *(const float2*)(qh + (size_t)(i0 + low) * HIDDEN + 4 * t + 2 * hi);

  float m_run = -1e30f, s_run = 0.f;
  v8f o1 = {}, o2 = {};

  for (int jt = 0; jt < 16; ++jt) {
    int jb = w * 256 + jt * 16;
    // T[j_l][i] = sum_d K[j][d] Q[i][d]
    v8f T = {};
#pragma unroll
    for (int t = 0; t < 8; ++t) {
      float2 ka = *(const float2*)(kh + (size_t)(jb + low) * HIDDEN + 4 * t + 2 * hi);
      T = wmma4(ka.x, ka.y, qB[t].x, qB[t].y, T);
    }
    float P[8];
    float tmax = -1e30f;
#pragma unroll
    for (int r = 0; r < 8; ++r) {
      float b = bh[(size_t)(jb + r + 8 * hi) * L_DIM + i0 + low];  // bias[h][j][i]
      float tv = T[r] + b;
      P[r] = tv;
      tmax = fmaxf(tmax, tv);
    }
    tmax = fmaxf(tmax, __shfl_xor(tmax, 16, 32));   // combine j halves per column i
    float nm = fmaxf(m_run, tmax);
    float corr = __expf(m_run - nm);
    m_run = nm;
    s_run *= corr;
#pragma unroll
    for (int r = 0; r < 8; ++r) {
      P[r] = __expf(P[r] - nm);
      s_run += P[r];
      o1[r] = o1[r] * corr;
      o2[r] = o2[r] * corr;
    }
    // O^T += V^T * P : B operand = P[j][i] relaid via wave shuffles.
#pragma unroll
    for (int t = 0; t < 4; ++t) {
      int src = low + ((t >= 2) ? 16 : 0);
      float pa = __shfl(P[(4 * t) & 7],     src, 32);
      float pb = __shfl(P[(4 * t + 2) & 7], src, 32);
      float pc = __shfl(P[(4 * t + 1) & 7], src, 32);
      float pd = __shfl(P[(4 * t + 3) & 7], src, 32);
      float B0 = hi ? pb : pa;
      float B1 = hi ? pd : pc;
      const float* vb = vh + (size_t)(jb + 4 * t + 2 * hi) * HIDDEN + low;
      float a10 = vb[0],  a11 = vb[HIDDEN];
      float a20 = vb[16], a21 = vb[HIDDEN + 16];
      o1 = wmma4(a10, a11, B0, B1, o1);
      o2 = wmma4(a20, a21, B0, B1, o2);
    }
  }

  s_run += __shfl_xor(s_run, 16, 32);
  if (l < 16) { m_s[w][low] = m_run; s_s[w][low] = s_run; }
#pragma unroll
  for (int r = 0; r < 8; ++r) {
    acc_s[w][r + 8 * hi][low]      = o1[r];
    acc_s[w][16 + r + 8 * hi][low] = o2[r];
  }
  __syncthreads();

  // Merge 4 partial online-softmax states, apply gate, emit [i][h*32+d].
  for (int idx = tid; idx < 512; idx += 128) {
    int i = idx & 15, d = idx >> 4;
    float M = fmaxf(fmaxf(m_s[0][i], m_s[1][i]), fmaxf(m_s[2][i], m_s[3][i]));
    float S = 0.f, O = 0.f;
#pragma unroll
    for (int ww = 0; ww < 4; ++ww) {
      float e = __expf(m_s[ww][i] - M);
      S += s_s[ww][i] * e;
      O += acc_s[ww][d][i] * e;
    }
    float gate = g[(size_t)h * L_DIM * HIDDEN + (size_t)(i0 + i) * HIDDEN + d];
    op[(size_t)(i0 + i) * SEQ_DIM + h * HIDDEN + d] = O * __builtin_amdgcn_rcpf(S) * gate;
  }
}

// ---------------------------------------------------------------------------
// 5. Output projection: out = op[1024,384] @ Wo^T + bo. grid=(64,24), 1 wave.
// ---------------------------------------------------------------------------
__global__ void out_proj_kernel(const float* __restrict__ A, const float* __restrict__ Wo,
                                const float* __restrict__ bo, float* __restrict__ out) {
  int i0 = blockIdx.x * 16, n0 = blockIdx.y * 16;
  v8f acc = gemm_tile_f32(A + (size_t)i0 * SEQ_DIM, SEQ_DIM,
                          Wo + (size_t)n0 * SEQ_DIM, SEQ_DIM, SEQ_DIM);
  int l = threadIdx.x & 31, low = l & 15, hi = l >> 4;
  int c = n0 + low;
  float bb = bo[c];
#pragma unroll
  for (int r = 0; r < 8; ++r) {
    int i = i0 + r + 8 * hi;
    out[(size_t)i * SEQ_DIM + c] = acc[r] + bb;
  }
}

// ---------------------------------------------------------------------------
extern "C" void kernel_launch(void* const* d_in, const int* in_sizes, int n_in,
                              void* d_out, int out_size, void* d_ws, size_t ws_size,
                              hipStream_t stream) {
  (void)in_sizes; (void)n_in; (void)out_size; (void)ws_size;
  const float* seq     = (const float*)d_in[0];
  const float* pair    = (const float*)d_in[1];
  const float* s_gamma = (const float*)d_in[2];
  const float* s_beta  = (const float*)d_in[3];
  const float* z_gamma = (const float*)d_in[4];
  const float* z_beta  = (const float*)d_in[5];
  const float* Wb      = (const float*)d_in[6];
  const float* Wq      = (const float*)d_in[7];
  const float* Wk      = (const float*)d_in[8];
  const float* Wv      = (const float*)d_in[9];
  const float* Wg      = (const float*)d_in[10];
  const float* bg      = (const float*)d_in[11];
  const float* Wo      = (const float*)d_in[12];
  const float* bo      = (const float*)d_in[13];
  float* out = (float*)d_out;

  // Workspace map (floats): ~59.8 MB total.
  float* ws     = (float*)d_ws;
  const size_t SN = (size_t)L_DIM * SEQ_DIM;            // 393216
  float* s_norm = ws;
  float* q      = s_norm + SN;
  float* k      = q + SN;
  float* v      = k + SN;
  float* gt     = v + SN;
  float* op     = gt + SN;
  float* bias   = op + SN;                               // [h][j][i], 12*1024*1024

  ln_seq_kernel<<<L_DIM, 32, 0, stream>>>(seq, s_gamma, s_beta, s_norm);
  proj_qkvg_kernel<<<dim3(L_DIM / 16, SEQ_DIM / 16, 4), 32, 0, stream>>>(
      s_norm, Wq, Wk, Wv, Wg, bg, q, k, v, gt);
  pair_bias_kernel<<<dim3(L_DIM / 8, L_DIM / 16), 256, 0, stream>>>(
      pair, z_gamma, z_beta, Wb, bias);
  attention_kernel<<<dim3(L_DIM / 16, HEADS), 128, 0, stream>>>(q, k, v, gt, bias, op);
  out_proj_kernel<<<dim3(L_DIM / 16, SEQ_DIM / 16), 32, 0, stream>>>(op, Wo, bo, out);
}